// CharRNN_80487687127332
// MI455X (gfx1250) — compile-verified
//
#include <hip/hip_runtime.h>

#define VOCAB   256
#define EMBED   32
#define HIDDEN  32
#define LAYERS  10
#define BATCH   1024
#define SEQT    1024

#define RS      34                    // padded LDS row stride (floats): conflict-free A loads
#define TILEF   (16 * RS)             // floats per [16 x H] buffer slice
#define NBUF    (LAYERS * 2 * TILEF)  // double-buffered per-layer activations
#define NEMB    (VOCAB * RS)

typedef __attribute__((ext_vector_type(2))) float v2f;
typedef __attribute__((ext_vector_type(8))) float v8f;

__device__ __forceinline__ v8f wmma_k4(v2f a, v2f b, v8f c) {
  // D(16x16,f32) = A(16x4,f32) * B(4x16,f32) + C
  return __builtin_amdgcn_wmma_f32_16x16x4_f32(false, a, false, b, (short)0, c,
                                               false, false);
}

__device__ __forceinline__ float dev_tanh(float x) {
#if __has_builtin(__builtin_amdgcn_tanhf)
  return __builtin_amdgcn_tanhf(x);              // native V_TANH_F32 on gfx1250
#else
  float ax = __builtin_fabsf(x);
  float e  = __expf(-2.0f * ax);
  float y  = __fdividef(1.0f - e, 1.0f + e);
  return __builtin_copysignf(y, x);
#endif
}

__global__ void __launch_bounds__(LAYERS * 32)
rnn_pipeline_kernel(const int* __restrict__ x, const float* __restrict__ emb,
                    const float* __restrict__ W_ih, const float* __restrict__ W_hh,
                    const float* __restrict__ b_ih, const float* __restrict__ b_hh,
                    const float* __restrict__ W_fc, const float* __restrict__ b_fc,
                    float* __restrict__ out) {
  extern __shared__ float smem[];
  float* buf  = smem;          // [LAYERS][2][16][RS] layer outputs (double buffered)
  float* embs = smem + NBUF;   // [VOCAB][RS] embedding table

  const int tid  = threadIdx.x;
  const int lane = tid & 31;
  const int l    = tid >> 5;                 // wave id == layer id
  const int R    = blockIdx.x * 16;          // batch row base for this workgroup

  // zero h-state buffers (t==0 reads must see 0), stage embedding table
  for (int i = tid; i < NBUF; i += LAYERS * 32) buf[i] = 0.0f;
  for (int i = tid; i < VOCAB * EMBED; i += LAYERS * 32)
    embs[(i >> 5) * RS + (i & 31)] = emb[i];

  const int m    = lane & 15;                // A: row M / B,D: col N
  const int koff = (lane >> 4) * 2;          // K sub-offset within a 4-block
  const int rb   = (lane >> 4) * 8;          // D: row base

  // resident weights in B-operand layout: B[k][n] = W[16c+n][k]
  v2f wih[2][8], whh[2][8];
  float bias[2];
  {
    const float* Wi = W_ih + l * HIDDEN * EMBED;
    const float* Wh = W_hh + l * HIDDEN * HIDDEN;
#pragma unroll
    for (int c = 0; c < 2; ++c) {
      const int row = c * 16 + m;
#pragma unroll
      for (int j = 0; j < 8; ++j) {
        wih[c][j] = *(const v2f*)(Wi + row * EMBED  + 4 * j + koff);
        whh[c][j] = *(const v2f*)(Wh + row * HIDDEN + 4 * j + koff);
      }
      bias[c] = b_ih[l * HIDDEN + row] + b_hh[l * HIDDEN + row];
    }
  }

  __syncthreads();

  // systolic pipeline: wave l processes time t = i - l at global iteration i
  for (int i = 0; i < SEQT + LAYERS - 1; ++i) {
    const int t = i - l;
    if ((unsigned)t < (unsigned)SEQT) {      // wave-uniform => EXEC stays all-ones
      const int pin  = (i - 1) & 1;
      const int pout = i & 1;

      v2f ah[8], ax[8];
      const float* hin = buf + (l * 2 + pin) * TILEF + m * RS + koff;
#pragma unroll
      for (int j = 0; j < 8; ++j) ah[j] = *(const v2f*)(hin + 4 * j);

      if (l == 0) {                          // input = embedding lookup
        const int tok   = x[(R + m) * SEQT + t];
        const float* xe = embs + tok * RS + koff;
#pragma unroll
        for (int j = 0; j < 8; ++j) ax[j] = *(const v2f*)(xe + 4 * j);
      } else {                               // input = layer l-1 output at same t
        const float* xin = buf + ((l - 1) * 2 + pin) * TILEF + m * RS + koff;
#pragma unroll
        for (int j = 0; j < 8; ++j) ax[j] = *(const v2f*)(xin + 4 * j);
      }

      v8f d0, d1;
#pragma unroll
      for (int k = 0; k < 8; ++k) { d0[k] = bias[0]; d1[k] = bias[1]; }
#pragma unroll
      for (int j = 0; j < 8; ++j) {          // x @ W_ih^T  (two 16-col tiles)
        d0 = wmma_k4(ax[j], wih[0][j], d0);
        d1 = wmma_k4(ax[j], wih[1][j], d1);
      }
#pragma unroll
      for (int j = 0; j < 8; ++j) {          // h @ W_hh^T
        d0 = wmma_k4(ah[j], whh[0][j], d0);
        d1 = wmma_k4(ah[j], whh[1][j], d1);
      }

      float* hout = buf + (l * 2 + pout) * TILEF;
#pragma unroll
      for (int r = 0; r < 8; ++r) {          // D: VGPR r -> (row rb+r, col m [+16])
        hout[(rb + r) * RS + m]      = dev_tanh(d0[r]);
        hout[(rb + r) * RS + m + 16] = dev_tanh(d1[r]);
      }
    }
    __syncthreads();
  }

  // final FC: out[16 x 256] = h_top @ W_fc^T + b_fc, vocab tiles split over waves
  const float* htop = buf + ((LAYERS - 1) * 2 + ((SEQT + LAYERS - 2) & 1)) * TILEF;
  v2f ahf[8];
#pragma unroll
  for (int j = 0; j < 8; ++j)
    ahf[j] = *(const v2f*)(htop + m * RS + 4 * j + koff);

  for (int ct = l; ct < VOCAB / 16; ct += LAYERS) {
    const int vrow = ct * 16 + m;
    v2f wb[8];
#pragma unroll
    for (int j = 0; j < 8; ++j)
      wb[j] = *(const v2f*)(W_fc + vrow * HIDDEN + 4 * j + koff);
    const float bf = b_fc[vrow];
    v8f d;
#pragma unroll
    for (int k = 0; k < 8; ++k) d[k] = bf;
#pragma unroll
    for (int j = 0; j < 8; ++j) d = wmma_k4(ahf[j], wb[j], d);
#pragma unroll
    for (int r = 0; r < 8; ++r)
      out[(R + rb + r) * VOCAB + vrow] = d[r];
  }
}

extern "C" void kernel_launch(void* const* d_in, const int* in_sizes, int n_in,
                              void* d_out, int out_size, void* d_ws, size_t ws_size,
                              hipStream_t stream) {
  (void)in_sizes; (void)n_in; (void)out_size; (void)d_ws; (void)ws_size;
  const int*   x   = (const int*)d_in[0];
  const float* emb = (const float*)d_in[1];
  const float* Wih = (const float*)d_in[2];
  const float* Whh = (const float*)d_in[3];
  const float* bih = (const float*)d_in[4];
  const float* bhh = (const float*)d_in[5];
  const float* Wfc = (const float*)d_in[6];
  const float* bfc = (const float*)d_in[7];
  float* out = (float*)d_out;

  const size_t shmem = (size_t)(NBUF + NEMB) * sizeof(float);  // ~76.5 KB dynamic LDS
  hipLaunchKernelGGL(rnn_pipeline_kernel, dim3(BATCH / 16), dim3(LAYERS * 32),
                     shmem, stream, x, emb, Wih, Whh, bih, bhh, Wfc, bfc, out);
}